// MMMambaEncoderLayer_78460462563603
// MI455X (gfx1250) — compile-verified
//
#include <hip/hip_runtime.h>
#include <math.h>

#define BB 8
#define LL 1024
#define DM 768
#define DI 1536
#define NS 16
#define MROWS (BB * LL)   // 8192

typedef float v2f __attribute__((ext_vector_type(2)));
typedef float v8f __attribute__((ext_vector_type(8)));

__device__ __forceinline__ float silu_f(float v) {
    return v / (1.0f + __expf(-v));
}

#define ACT_NONE 0
#define ACT_SOFTPLUS 1

// ---------------------------------------------------------------------------
// Generic fp32 WMMA GEMM: C[m,n] = act( sum_k A[m*lda+k] * W[n*ldw+k] + bias[n] )
// Requirements: M % 128 == 0, K % 16 == 0. N arbitrary (guarded).
// Block tile 128x128, 256 threads = 8 waves (4 M-groups x 2 N-groups).
// Each wave computes a 32x64 strip = eight 16x16 D tiles, so per K=4 step it
// issues 8 independent V_WMMA_F32_16X16X4_F32 against 6 LDS fragment fetches.
// ---------------------------------------------------------------------------
__launch_bounds__(256)
__global__ void gemm_wmma_f32(const float* __restrict__ A, int lda,
                              const float* __restrict__ W, int ldw,
                              float* __restrict__ C, int ldc,
                              int M, int N, int K,
                              const float* __restrict__ bias, int act)
{
    __shared__ float As[128][17];
    __shared__ float Ws[128][17];

    const int tid  = threadIdx.x;
    const int wave = tid >> 5;
    const int lane = tid & 31;
    const int hi   = (lane >> 4) & 1;   // lanes 16-31 hold K+2/K+3 halves
    const int l15  = lane & 15;
    const int wm   = wave & 3;          // M group: rows wm*32 .. wm*32+31
    const int wn   = wave >> 2;         // N group: cols wn*64 .. wn*64+63

    const int rowBase = blockIdx.y * 128;
    const int colBase = blockIdx.x * 128;

    v8f acc[2][4];
    #pragma unroll
    for (int i = 0; i < 2; ++i)
        #pragma unroll
        for (int j = 0; j < 4; ++j)
            acc[i][j] = (v8f){};

    // Global->LDS staging: each thread loads two float4 of A and two of W.
    const int lrow = tid >> 1;          // 0..127
    const int lk   = (tid & 1) * 8;     // 0 or 8

    const float* aSrcBase = A + (size_t)(rowBase + lrow) * lda + lk;
    const int    ncol     = colBase + lrow;
    const float* wSrcBase = W + (size_t)ncol * ldw + lk;

    for (int k0 = 0; k0 < K; k0 += 16) {
        {   // A tile: 128 rows x 16 k
            float4 v0 = *(const float4*)(aSrcBase + k0);
            float4 v1 = *(const float4*)(aSrcBase + k0 + 4);
            As[lrow][lk + 0] = v0.x; As[lrow][lk + 1] = v0.y;
            As[lrow][lk + 2] = v0.z; As[lrow][lk + 3] = v0.w;
            As[lrow][lk + 4] = v1.x; As[lrow][lk + 5] = v1.y;
            As[lrow][lk + 6] = v1.z; As[lrow][lk + 7] = v1.w;
        }
        {   // W tile: 128 cols x 16 k (zero-pad past N)
            if (ncol < N) {
                float4 v0 = *(const float4*)(wSrcBase + k0);
                float4 v1 = *(const float4*)(wSrcBase + k0 + 4);
                Ws[lrow][lk + 0] = v0.x; Ws[lrow][lk + 1] = v0.y;
                Ws[lrow][lk + 2] = v0.z; Ws[lrow][lk + 3] = v0.w;
                Ws[lrow][lk + 4] = v1.x; Ws[lrow][lk + 5] = v1.y;
                Ws[lrow][lk + 6] = v1.z; Ws[lrow][lk + 7] = v1.w;
            } else {
                #pragma unroll
                for (int q = 0; q < 8; ++q) Ws[lrow][lk + q] = 0.f;
            }
        }
        // Prefetch next K chunk into cache while this chunk computes.
        if (k0 + 16 < K) {
            __builtin_prefetch(aSrcBase + k0 + 16, 0, 3);
            if (ncol < N) __builtin_prefetch(wSrcBase + k0 + 16, 0, 3);
        }
        __syncthreads();

        #pragma unroll
        for (int kk = 0; kk < 16; kk += 4) {
            const int kA = kk + (hi ? 2 : 0);
            // Two A fragments (rows wm*32 and wm*32+16)
            v2f a0, a1;
            a0.x = As[wm * 32 +      l15][kA];
            a0.y = As[wm * 32 +      l15][kA + 1];
            a1.x = As[wm * 32 + 16 + l15][kA];
            a1.y = As[wm * 32 + 16 + l15][kA + 1];
            // Four B fragments (cols wn*64 + 16*j)
            v2f b[4];
            #pragma unroll
            for (int j = 0; j < 4; ++j) {
                b[j].x = Ws[wn * 64 + 16 * j + l15][kA];
                b[j].y = Ws[wn * 64 + 16 * j + l15][kA + 1];
            }
            #pragma unroll
            for (int j = 0; j < 4; ++j) {
                acc[0][j] = __builtin_amdgcn_wmma_f32_16x16x4_f32(
                    false, a0, false, b[j], (short)0, acc[0][j], false, false);
                acc[1][j] = __builtin_amdgcn_wmma_f32_16x16x4_f32(
                    false, a1, false, b[j], (short)0, acc[1][j], false, false);
            }
        }
        __syncthreads();
    }

    // D layout: VGPR v, lanes 0-15 -> row v, lanes 16-31 -> row v+8; col = l15
    #pragma unroll
    for (int i = 0; i < 2; ++i) {
        const int row0 = rowBase + wm * 32 + i * 16 + (hi ? 8 : 0);
        #pragma unroll
        for (int j = 0; j < 4; ++j) {
            const int col = colBase + wn * 64 + 16 * j + l15;
            if (col < N) {
                const float bval = bias ? bias[col] : 0.f;
                #pragma unroll
                for (int v = 0; v < 8; ++v) {
                    float r = acc[i][j][v] + bval;
                    if (act == ACT_SOFTPLUS)
                        r = (r > 20.f) ? r : log1pf(__expf(r));
                    C[(size_t)(row0 + v) * ldc + col] = r;
                }
            }
        }
    }
}

// ---------------------------------------------------------------------------
// Depthwise causal conv (fwd) + anti-causal conv (bwd) + SiLU, fused.
// xz holds [xi | z] rows of 3072; we read xi (cols 0..1535).
// xf[l] = silu(sum_j cwf[j]*xi[l-3+j] + cbf)
// xb[l] = silu(sum_j cwb[j]*xi[l+3-j] + cbb)   (== causal conv on reversed seq)
// ---------------------------------------------------------------------------
__launch_bounds__(256)
__global__ void conv_silu_kernel(const float* __restrict__ xz,
                                 const float* __restrict__ cwf, const float* __restrict__ cbf,
                                 const float* __restrict__ cwb, const float* __restrict__ cbb,
                                 float* __restrict__ xf, float* __restrict__ xb)
{
    const size_t idx = (size_t)blockIdx.x * blockDim.x + threadIdx.x;
    if (idx >= (size_t)MROWS * DI) return;
    const int d   = (int)(idx % DI);
    const int row = (int)(idx / DI);
    const int l   = row % LL;
    const int b   = row / LL;

    float accf = cbf[d];
    float accb = cbb[d];
    #pragma unroll
    for (int j = 0; j < 4; ++j) {
        const int lf = l - 3 + j;
        if (lf >= 0)
            accf += cwf[d * 4 + j] * xz[(size_t)(b * LL + lf) * 3072 + d];
        const int lb = l + 3 - j;
        if (lb < LL)
            accb += cwb[d * 4 + j] * xz[(size_t)(b * LL + lb) * 3072 + d];
    }
    xf[idx] = silu_f(accf);
    xb[idx] = silu_f(accb);
}

// ---------------------------------------------------------------------------
// Selective scan: one thread per (b, d) channel, h[16] in registers.
// dbl rows of 80: [dt_rank(48) | B(16) | C(16)].
// reverse=1 walks l = L-1..0 (bwd direction, natural-order output).
// accumulate=1 adds into y (bwd adds onto fwd result).
// ---------------------------------------------------------------------------
__launch_bounds__(256)
__global__ void scan_kernel(const float* __restrict__ u,
                            const float* __restrict__ dt,
                            const float* __restrict__ dbl,
                            const float* __restrict__ A_log,
                            const float* __restrict__ Dp,
                            float* __restrict__ y,
                            int reverse, int accumulate)
{
    const int idx = blockIdx.x * blockDim.x + threadIdx.x;
    if (idx >= BB * DI) return;
    const int d = idx % DI;
    const int b = idx / DI;

    float Acoef[NS];
    float h[NS];
    #pragma unroll
    for (int n = 0; n < NS; ++n) {
        Acoef[n] = -__expf(A_log[d * NS + n]);
        h[n] = 0.f;
    }
    const float Dd = Dp[d];

    for (int t = 0; t < LL; ++t) {
        const int l = reverse ? (LL - 1 - t) : t;
        const size_t base = (size_t)(b * LL + l);
        const float u_  = u[base * DI + d];
        const float dt_ = dt[base * DI + d];
        const float* bc = dbl + base * 80;
        float yv = 0.f;
        #pragma unroll
        for (int n = 0; n < NS; ++n) {
            const float Bv = bc[48 + n];
            const float Cv = bc[64 + n];
            const float dA = __expf(dt_ * Acoef[n]);
            h[n] = h[n] * dA + dt_ * Bv * u_;
            yv += h[n] * Cv;
        }
        yv += u_ * Dd;
        const size_t o = base * DI + d;
        if (accumulate) y[o] += yv;
        else            y[o] = yv;
    }
}

// y = (yf + yb) * silu(z);  z is cols 1536..3071 of xz rows
__launch_bounds__(256)
__global__ void gate_kernel(const float* __restrict__ xz, float* __restrict__ y)
{
    const size_t idx = (size_t)blockIdx.x * blockDim.x + threadIdx.x;
    if (idx >= (size_t)MROWS * DI) return;
    const int d   = (int)(idx % DI);
    const int row = (int)(idx / DI);
    const float z = xz[(size_t)row * 3072 + DI + d];
    y[idx] *= silu_f(z);
}

// out = x_in + rmsnorm(o1, w); one block per 768-row
__launch_bounds__(256)
__global__ void rmsnorm_res_kernel(const float* __restrict__ o1,
                                   const float* __restrict__ xin,
                                   const float* __restrict__ w,
                                   float* __restrict__ out)
{
    __shared__ float red[256];
    const int row = blockIdx.x;
    const int tid = threadIdx.x;
    const float* src = o1 + (size_t)row * DM;

    float ss = 0.f;
    float v[3];
    #pragma unroll
    for (int i = 0; i < 3; ++i) {
        v[i] = src[tid + i * 256];
        ss += v[i] * v[i];
    }
    red[tid] = ss;
    __syncthreads();
    for (int s = 128; s > 0; s >>= 1) {
        if (tid < s) red[tid] += red[tid + s];
        __syncthreads();
    }
    const float scale = rsqrtf(red[0] / (float)DM + 1e-5f);
    #pragma unroll
    for (int i = 0; i < 3; ++i) {
        const int e = tid + i * 256;
        out[(size_t)row * DM + e] = xin[(size_t)row * DM + e] + v[i] * scale * w[e];
    }
}

// ---------------------------------------------------------------------------
extern "C" void kernel_launch(void* const* d_in, const int* in_sizes, int n_in,
                              void* d_out, int out_size, void* d_ws, size_t ws_size,
                              hipStream_t stream)
{
    const float* m_x         = (const float*)d_in[0];
    const float* n_x         = (const float*)d_in[1];
    const float* in_proj_w   = (const float*)d_in[2];
    const float* conv_w      = (const float*)d_in[3];
    const float* conv_b      = (const float*)d_in[4];
    const float* x_proj_w    = (const float*)d_in[5];
    const float* dt_proj_w   = (const float*)d_in[6];
    const float* dt_proj_b   = (const float*)d_in[7];
    const float* A_log       = (const float*)d_in[8];
    const float* Dp          = (const float*)d_in[9];
    const float* conv_w_b    = (const float*)d_in[10];
    const float* conv_b_b    = (const float*)d_in[11];
    const float* x_proj_w_b  = (const float*)d_in[12];
    const float* dt_proj_w_b = (const float*)d_in[13];
    const float* dt_proj_b_b = (const float*)d_in[14];
    const float* A_log_b     = (const float*)d_in[15];
    const float* D_b         = (const float*)d_in[16];
    const float* out_proj_w  = (const float*)d_in[17];
    const float* norm1_w     = (const float*)d_in[18];
    const float* norm2_w     = (const float*)d_in[19];

    float* ws = (float*)d_ws;
    size_t off = 0;
    float* xz  = ws + off; off += (size_t)MROWS * 3072;
    float* xf  = ws + off; off += (size_t)MROWS * DI;
    float* xb  = ws + off; off += (size_t)MROWS * DI;
    float* dbl = ws + off; off += (size_t)MROWS * 80;
    float* dtb = ws + off; off += (size_t)MROWS * DI;
    float* yb  = ws + off; off += (size_t)MROWS * DI;
    float* o1  = ws + off; off += (size_t)MROWS * DM;

    const int elemBlocks = (int)(((size_t)MROWS * DI + 255) / 256);
    const int scanBlocks = (BB * DI + 255) / 256;

    for (int s = 0; s < 2; ++s) {
        const float* xin = s ? n_x : m_x;
        const float* nw  = s ? norm2_w : norm1_w;
        float* outp = (float*)d_out + (size_t)s * MROWS * DM;

        // 1) in_proj: xz = xin @ Win^T   (M=8192, N=3072, K=768)
        {
            dim3 g((3072 + 127) / 128, MROWS / 128);
            gemm_wmma_f32<<<g, 256, 0, stream>>>(xin, DM, in_proj_w, DM,
                                                 xz, 3072, MROWS, 3072, DM,
                                                 nullptr, ACT_NONE);
        }
        // 2) depthwise conv (both directions) + silu
        conv_silu_kernel<<<elemBlocks, 256, 0, stream>>>(
            xz, conv_w, conv_b, conv_w_b, conv_b_b, xf, xb);

        // ---- forward direction ----
        {   // x_proj: dbl = xf @ xpw^T  (N=80, K=1536)
            dim3 g((80 + 127) / 128, MROWS / 128);
            gemm_wmma_f32<<<g, 256, 0, stream>>>(xf, DI, x_proj_w, DI,
                                                 dbl, 80, MROWS, 80, DI,
                                                 nullptr, ACT_NONE);
        }
        {   // dt = softplus(dbl[:, :48] @ dtw^T + bias)  (N=1536, K=48)
            dim3 g((DI + 127) / 128, MROWS / 128);
            gemm_wmma_f32<<<g, 256, 0, stream>>>(dbl, 80, dt_proj_w, 48,
                                                 dtb, DI, MROWS, DI, 48,
                                                 dt_proj_b, ACT_SOFTPLUS);
        }
        scan_kernel<<<scanBlocks, 256, 0, stream>>>(
            xf, dtb, dbl, A_log, Dp, yb, /*reverse=*/0, /*accumulate=*/0);

        // ---- backward direction ----
        {
            dim3 g((80 + 127) / 128, MROWS / 128);
            gemm_wmma_f32<<<g, 256, 0, stream>>>(xb, DI, x_proj_w_b, DI,
                                                 dbl, 80, MROWS, 80, DI,
                                                 nullptr, ACT_NONE);
        }
        {
            dim3 g((DI + 127) / 128, MROWS / 128);
            gemm_wmma_f32<<<g, 256, 0, stream>>>(dbl, 80, dt_proj_w_b, 48,
                                                 dtb, DI, MROWS, DI, 48,
                                                 dt_proj_b_b, ACT_SOFTPLUS);
        }
        scan_kernel<<<scanBlocks, 256, 0, stream>>>(
            xb, dtb, dbl, A_log_b, D_b, yb, /*reverse=*/1, /*accumulate=*/1);

        // gate: y = (yf + yb) * silu(z)
        gate_kernel<<<elemBlocks, 256, 0, stream>>>(xz, yb);

        // out_proj: o1 = y @ Wout^T  (N=768, K=1536)
        {
            dim3 g((DM + 127) / 128, MROWS / 128);
            gemm_wmma_f32<<<g, 256, 0, stream>>>(yb, DI, out_proj_w, DI,
                                                 o1, DM, MROWS, DM, DI,
                                                 nullptr, ACT_NONE);
        }
        // residual + rmsnorm
        rmsnorm_res_kernel<<<MROWS, 256, 0, stream>>>(o1, xin, nw, outp);
    }
}